// KMaxPool1d_50070728737615
// MI455X (gfx1250) — compile-verified
//
#include <hip/hip_runtime.h>

#define ROW_L 4096
#define THREADS 256
#define PER_THREAD (ROW_L / THREADS) /* 16 */
#define KSEL 8

__device__ __forceinline__ unsigned monoKey(float f) {
  unsigned u = __float_as_uint(f);
  unsigned mask = ((unsigned)(((int)u) >> 31)) | 0x80000000u;
  return u ^ mask; // order-preserving fp32 -> u32 (bijective)
}

__device__ __forceinline__ float invKey(unsigned k) {
  unsigned mask = ((unsigned)(((int)(~k)) >> 31)) | 0x80000000u;
  return __uint_as_float(k ^ mask); // exact inverse of monoKey
}

#define CE_DESC(a, b)                         \
  {                                           \
    unsigned hi_ = (a) > (b) ? (a) : (b);     \
    unsigned lo_ = (a) > (b) ? (b) : (a);     \
    (a) = hi_;                                \
    (b) = lo_;                                \
  }

#define CE_ASC(a, b)                          \
  {                                           \
    unsigned hi_ = (a) > (b) ? (a) : (b);     \
    unsigned lo_ = (a) > (b) ? (b) : (a);     \
    (a) = lo_;                                \
    (b) = hi_;                                \
  }

// Bitonic sort of 8 values, descending. 24 compare-exchanges.
__device__ __forceinline__ void sort8_desc(unsigned x[KSEL]) {
  // sort4 desc on x0..x3 (desc pair + asc pair -> bitonic, merge desc)
  CE_DESC(x[0], x[1]); CE_ASC(x[2], x[3]);
  CE_DESC(x[0], x[2]); CE_DESC(x[1], x[3]);
  CE_DESC(x[0], x[1]); CE_DESC(x[2], x[3]);
  // sort4 asc on x4..x7
  CE_ASC(x[4], x[5]); CE_DESC(x[6], x[7]);
  CE_ASC(x[4], x[6]); CE_ASC(x[5], x[7]);
  CE_ASC(x[4], x[5]); CE_ASC(x[6], x[7]);
  // x0..x3 desc + x4..x7 asc = bitonic 8; merge desc (dist 4,2,1)
  CE_DESC(x[0], x[4]); CE_DESC(x[1], x[5]); CE_DESC(x[2], x[6]); CE_DESC(x[3], x[7]);
  CE_DESC(x[0], x[2]); CE_DESC(x[1], x[3]); CE_DESC(x[4], x[6]); CE_DESC(x[5], x[7]);
  CE_DESC(x[0], x[1]); CE_DESC(x[2], x[3]); CE_DESC(x[4], x[5]); CE_DESC(x[6], x[7]);
}

// Given a[], b[] sorted desc, write top-8 of union into a[], sorted desc.
__device__ __forceinline__ void mergeTop8(unsigned a[KSEL], const unsigned b[KSEL]) {
#pragma unroll
  for (int i = 0; i < KSEL; ++i) { // bitonic first stage: keep upper half
    unsigned x = a[i], y = b[KSEL - 1 - i];
    a[i] = x > y ? x : y;
  }
  // resort bitonic 8-sequence descending: stages 4, 2, 1 (12 CEs)
  CE_DESC(a[0], a[4]); CE_DESC(a[1], a[5]); CE_DESC(a[2], a[6]); CE_DESC(a[3], a[7]);
  CE_DESC(a[0], a[2]); CE_DESC(a[1], a[3]); CE_DESC(a[4], a[6]); CE_DESC(a[5], a[7]);
  CE_DESC(a[0], a[1]); CE_DESC(a[2], a[3]); CE_DESC(a[4], a[5]); CE_DESC(a[6], a[7]);
}

__global__ __launch_bounds__(THREADS)
void topk_rows_kernel(const float* __restrict__ in, float* __restrict__ out) {
  __shared__ float rowBuf[ROW_L];               // 16 KB: staged row
  __shared__ unsigned lists[THREADS][KSEL];     // 8 KB : per-thread top-8 key lists
  __shared__ unsigned scanBuf[THREADS];         // 1 KB : packed (gt,eq) scan

  const int tid = threadIdx.x;
  const long long row = blockIdx.x;
  const float* rowPtr = in + row * (long long)ROW_L;

  // ------------- Stage row global -> LDS via CDNA5 async copy (stream, NT) ------
  {
    unsigned ldsBase = (unsigned)(size_t)(&rowBuf[0]); // low 32 bits = LDS offset
    unsigned long long gBase = (unsigned long long)(const void*)rowPtr;
#pragma unroll
    for (int it = 0; it < (ROW_L * 4) / (THREADS * 16); ++it) {
      unsigned byteOff = (unsigned)((it * THREADS + tid) * 16);
      unsigned ldsAddr = ldsBase + byteOff;
      unsigned long long gAddr = gBase + byteOff;
      asm volatile("global_load_async_to_lds_b128 %0, %1, off th:TH_LOAD_NT"
                   :
                   : "v"(ldsAddr), "v"(gAddr)
                   : "memory");
    }
    asm volatile("s_wait_asynccnt 0" ::: "memory");
  }
  __syncthreads();

  // ------------- Phase 1a: keys once, truncated bitonic top-8 selection ---------
  unsigned keys[PER_THREAD]; // original index order, kept for phase 2
  {
    const float4* rb = reinterpret_cast<const float4*>(&rowBuf[tid * PER_THREAD]);
#pragma unroll
    for (int q = 0; q < PER_THREAD / 4; ++q) {
      float4 v4 = rb[q];
      keys[q * 4 + 0] = monoKey(v4.x);
      keys[q * 4 + 1] = monoKey(v4.y);
      keys[q * 4 + 2] = monoKey(v4.z);
      keys[q * 4 + 3] = monoKey(v4.w);
    }
  }

  unsigned m[KSEL], b[KSEL];
#pragma unroll
  for (int i = 0; i < KSEL; ++i) { m[i] = keys[i]; b[i] = keys[KSEL + i]; }
  sort8_desc(m);
  sort8_desc(b);
  mergeTop8(m, b); // m = sorted-desc top-8 of this thread's 16 keys

  // ------------- Phase 1b: bitonic merge tree over 256 sorted lists -------------
#pragma unroll
  for (int i = 0; i < KSEL; ++i) lists[tid][i] = m[i];

  for (int d = 1; d < THREADS; d <<= 1) {
    __syncthreads();
    if ((tid & (2 * d - 1)) == 0) {
      unsigned o[KSEL];
#pragma unroll
      for (int i = 0; i < KSEL; ++i) o[i] = lists[tid + d][i];
      mergeTop8(m, o);
#pragma unroll
      for (int i = 0; i < KSEL; ++i) lists[tid][i] = m[i];
    }
  }
  __syncthreads();
  const unsigned T = lists[0][KSEL - 1]; // key of row's 8th-largest value

  // ------------- Phase 2a: counts + block scan (index order) --------------------
  unsigned cnt_gt = 0, cnt_eq = 0;
#pragma unroll
  for (int j = 0; j < PER_THREAD; ++j) {
    cnt_gt += (keys[j] > T) ? 1u : 0u;
    cnt_eq += (keys[j] == T) ? 1u : 0u;
  }
  unsigned packed = (cnt_gt << 16) | cnt_eq;
  scanBuf[tid] = packed;
  __syncthreads();

  unsigned acc = packed;
  for (int off = 1; off < THREADS; off <<= 1) {
    unsigned add = (tid >= off) ? scanBuf[tid - off] : 0u;
    __syncthreads();
    acc += add;
    scanBuf[tid] = acc;
    __syncthreads();
  }
  const unsigned total = scanBuf[THREADS - 1];
  const unsigned ex = acc - packed; // exclusive prefix for this thread
  const unsigned total_gt = total >> 16;
  const unsigned need_eq = (unsigned)(KSEL - (int)total_gt); // >= 1 always

  // ------------- Phase 2b: emit selected values in original index order ---------
  float* outRow = out + row * (long long)KSEL;
  unsigned g = ex >> 16;      // running count of >T before this element
  unsigned e = ex & 0xFFFFu;  // running count of ==T before this element
#pragma unroll
  for (int j = 0; j < PER_THREAD; ++j) {
    unsigned k = keys[j];
    bool isgt = (k > T);
    bool iseq = (k == T);
    unsigned e_sel = e < need_eq ? e : need_eq; // selected equals so far
    if (isgt) {
      outRow[g + e_sel] = invKey(k);
    } else if (iseq && (e < need_eq)) {
      outRow[g + e] = invKey(k);
    }
    g += isgt ? 1u : 0u;
    e += iseq ? 1u : 0u;
  }
}

extern "C" void kernel_launch(void* const* d_in, const int* in_sizes, int n_in,
                              void* d_out, int out_size, void* d_ws, size_t ws_size,
                              hipStream_t stream) {
  (void)n_in; (void)d_ws; (void)ws_size; (void)out_size;
  const float* in = (const float*)d_in[0];
  float* out = (float*)d_out;
  const int rows = in_sizes[0] / ROW_L; // 32*512 = 16384
  topk_rows_kernel<<<rows, THREADS, 0, stream>>>(in, out);
}